// SpectralConv2d_fast_LoRA_GatedMoE_40613210751001
// MI455X (gfx1250) — compile-verified
//
#include <hip/hip_runtime.h>
#include <math.h>

typedef __attribute__((ext_vector_type(2))) float v2f;
typedef __attribute__((ext_vector_type(8))) float v8f;

__device__ __forceinline__ v8f wmma4(v2f a, v2f b, v8f c) {
  // D = A(16x4,f32) * B(4x16,f32) + C(16x16,f32)
  return __builtin_amdgcn_wmma_f32_16x16x4_f32(false, a, false, b, (short)0, c, false, false);
}

// ---------------- workspace layout (float offsets) ----------------
enum : size_t {
  OFF_FWC  = 0,                          // [256w][64ky]  cos(-2pi w ky/256) = cos
  OFF_FWS  = OFF_FWC + 256 * 64,         // [256w][64ky]  -sin
  OFF_FHC  = OFF_FWS + 256 * 64,         // [128r][256h]  cos(2pi kx h/256)
  OFF_FHS  = OFF_FHC + 128 * 256,        // [128r][256h]  +sin
  OFF_FHMS = OFF_FHS + 128 * 256,        // [128r][256h]  -sin
  OFF_GHC  = OFF_FHMS + 128 * 256,       // [256h][128r]  cos
  OFF_GHS  = OFF_GHC + 256 * 128,        // [256h][128r]  +sin
  OFF_GHMS = OFF_GHS + 256 * 128,        // [256h][128r]  -sin
  OFF_GWR  = OFF_GHMS + 256 * 128,       // [64ky][256w]  c_ky*cos/65536
  OFF_GWI  = OFF_GWR + 64 * 256,         // [64ky][256w] -c_ky*sin/65536
  OFF_TRE  = OFF_GWI + 64 * 256,         // [512 img][256h][64ky]  (also reused as U_re)
  OFF_TIM  = OFF_TRE + (size_t)512 * 256 * 64,
  OFF_XRE  = OFF_TIM + (size_t)512 * 256 * 64,   // [512][128r][64ky]
  OFF_XIM  = OFF_XRE + (size_t)512 * 128 * 64,
  OFF_YRE  = OFF_XIM + (size_t)512 * 128 * 64,   // [512][128r][64ky]
  OFF_YIM  = OFF_YRE + (size_t)512 * 128 * 64,
  OFF_END  = OFF_YIM + (size_t)512 * 128 * 64
};

// ---------------- twiddle init ----------------
__global__ __launch_bounds__(256) void k_twiddle(float* __restrict__ ws) {
  int i = blockIdx.x * 256 + threadIdx.x;          // 0..32767
  const float C = 6.283185307179586f / 256.0f;
  if (i < 16384) {
    // forward W-DFT: theta = 2pi*w*ky/256, e^{-i th} = cos - i sin
    int w = i >> 6, ky = i & 63;
    float th = C * (float)((w * ky) & 255);
    ws[OFF_FWC + i] = cosf(th);
    ws[OFF_FWS + i] = -sinf(th);
    // inverse W (irfft fold): y = Ure@Gwr + Uim@Gwi, scale 1/(H*W), c0=1 else 2
    int ky2 = i >> 8, w2 = i & 255;
    float th2 = C * (float)((ky2 * w2) & 255);
    float cc = (ky2 == 0 ? 1.0f : 2.0f) * (1.0f / 65536.0f);
    ws[OFF_GWR + i] = cc * cosf(th2);
    ws[OFF_GWI + i] = -cc * sinf(th2);
  }
  if (i < 32768) {
    // forward H-DFT: rows r: 0..63 -> kx=r (top), 64..127 -> kx=r+128 (bottom)
    int m = i >> 8, h = i & 255;
    int kx = (m < 64) ? m : m + 128;
    float th = C * (float)((kx * h) & 255);
    float c = cosf(th), s = sinf(th);
    ws[OFF_FHC + i] = c; ws[OFF_FHS + i] = s; ws[OFF_FHMS + i] = -s;
    // inverse H: [h][r]
    int h2 = i >> 7, m2 = i & 127;
    int kx2 = (m2 < 64) ? m2 : m2 + 128;
    float th2 = C * (float)((kx2 * h2) & 255);
    float c2 = cosf(th2), s2 = sinf(th2);
    ws[OFF_GHC + i] = c2; ws[OFF_GHS + i] = s2; ws[OFF_GHMS + i] = -s2;
  }
}

// ---------------- stage F1: row DFT along W (real input) ----------------
// T[img][h][ky] = sum_w x[img][h][w] * e^{-2pi i w ky/256}, ky=0..63
__global__ __launch_bounds__(256) void k_dft_w(const float* __restrict__ x,
                                               const float* __restrict__ fwc,
                                               const float* __restrict__ fws,
                                               float* __restrict__ tre,
                                               float* __restrict__ tim) {
  int img  = blockIdx.x;                       // 0..511 (b*64+ci)
  int tile = blockIdx.y * 8 + (threadIdx.x >> 5); // 0..63
  int ht = tile >> 2, kyt = tile & 3;
  int lane = threadIdx.x & 31;
  int ln = lane & 15;
  int kh = (lane >> 4) << 1;                   // K offset 0 or 2
  const float* xrow = x + (size_t)img * 65536 + (size_t)(ht * 16 + ln) * 256;
  int n = kyt * 16 + ln;
  v8f accr = {0,0,0,0,0,0,0,0}, acci = {0,0,0,0,0,0,0,0};
  for (int k0 = 0; k0 < 256; k0 += 4) {
    __builtin_prefetch(xrow + k0 + 64, 0, 0);
    v2f a;  a.x  = xrow[k0 + kh];               a.y  = xrow[k0 + kh + 1];
    v2f bc; bc.x = fwc[(k0 + kh) * 64 + n];     bc.y = fwc[(k0 + kh + 1) * 64 + n];
    v2f bs; bs.x = fws[(k0 + kh) * 64 + n];     bs.y = fws[(k0 + kh + 1) * 64 + n];
    accr = wmma4(a, bc, accr);
    acci = wmma4(a, bs, acci);
  }
  size_t base = (size_t)img * (256 * 64);
  int rowoff = ht * 16 + ((lane >> 4) << 3);
  for (int j = 0; j < 8; ++j) {
    tre[base + (size_t)(rowoff + j) * 64 + n] = accr[j];
    tim[base + (size_t)(rowoff + j) * 64 + n] = acci[j];
  }
}

// ---------------- stage F2: col DFT along H (complex) ----------------
// X[r][ky] = sum_h e^{-2pi i kx h/256} T[h][ky];  Xre = cos@Tre + sin@Tim ; Xim = cos@Tim - sin@Tre
__global__ __launch_bounds__(256) void k_dft_h(const float* __restrict__ tre,
                                               const float* __restrict__ tim,
                                               const float* __restrict__ fhc,
                                               const float* __restrict__ fhs,
                                               const float* __restrict__ fhms,
                                               float* __restrict__ xre,
                                               float* __restrict__ xim) {
  int img  = blockIdx.x;
  int tile = blockIdx.y * 8 + (threadIdx.x >> 5); // 0..31
  int rt = tile >> 2, kyt = tile & 3;
  int lane = threadIdx.x & 31;
  int ln = lane & 15;
  int kh = (lane >> 4) << 1;
  int m = rt * 16 + ln;
  int n = kyt * 16 + ln;
  size_t tb = (size_t)img * 16384;
  v8f accr = {0,0,0,0,0,0,0,0}, acci = {0,0,0,0,0,0,0,0};
  for (int k0 = 0; k0 < 256; k0 += 4) {
    v2f ac;  ac.x  = fhc[m * 256 + k0 + kh];   ac.y  = fhc[m * 256 + k0 + kh + 1];
    v2f as;  as.x  = fhs[m * 256 + k0 + kh];   as.y  = fhs[m * 256 + k0 + kh + 1];
    v2f ams; ams.x = fhms[m * 256 + k0 + kh];  ams.y = fhms[m * 256 + k0 + kh + 1];
    v2f br;  br.x  = tre[tb + (size_t)(k0 + kh) * 64 + n]; br.y = tre[tb + (size_t)(k0 + kh + 1) * 64 + n];
    v2f bi;  bi.x  = tim[tb + (size_t)(k0 + kh) * 64 + n]; bi.y = tim[tb + (size_t)(k0 + kh + 1) * 64 + n];
    accr = wmma4(ac, br, accr);
    accr = wmma4(as, bi, accr);
    acci = wmma4(ac, bi, acci);
    acci = wmma4(ams, br, acci);
  }
  size_t xb = (size_t)img * 8192;
  int rowoff = rt * 16 + ((lane >> 4) << 3);
  for (int j = 0; j < 8; ++j) {
    xre[xb + (size_t)(rowoff + j) * 64 + n] = accr[j];
    xim[xb + (size_t)(rowoff + j) * 64 + n] = acci[j];
  }
}

// ---------------- mixing: base tiles (dense CIxCO complex matmul per position) ----------------
__global__ __launch_bounds__(256) void k_mix_base(const float* __restrict__ xre,
                                                  const float* __restrict__ xim,
                                                  const float* __restrict__ w1re,
                                                  const float* __restrict__ w1im,
                                                  const float* __restrict__ w2re,
                                                  const float* __restrict__ w2im,
                                                  float* __restrict__ yre,
                                                  float* __restrict__ yim) {
  int pos = blockIdx.x;          // 0..255
  int corner = blockIdx.y;       // 0 = top corner cell (0,0), 1 = bottom cell (3,0)
  int X = pos >> 4, Y = pos & 15;
  int rio = corner ? (112 + X) : X;   // same row for input and output in r-space
  const float* wre = corner ? w2re : w1re;
  const float* wim = corner ? w2im : w1im;
  __shared__ float sxr[64][8], sxi[64][8];
  __shared__ float swr[64][64], swi[64][64];
  int t = threadIdx.x;
  for (int q = t; q < 512; q += 256) {
    int i = q >> 3, b = q & 7;
    size_t src = ((size_t)(b * 64 + i) * 128 + rio) * 64 + Y;
    sxr[i][b] = xre[src];
    sxi[i][b] = xim[src];
  }
  for (int q = t; q < 4096; q += 256) {
    int i = q >> 6, o = q & 63;
    size_t src = ((size_t)(i * 64 + o) * 16 + X) * 16 + Y;
    swr[i][o] = wre[src];
    swi[i][o] = wim[src];
  }
  __syncthreads();
  for (int p = t; p < 512; p += 256) {
    int b = p >> 6, o = p & 63;
    float ar = 0.f, ai = 0.f;
    for (int i = 0; i < 64; ++i) {
      float xr = sxr[i][b], xi = sxi[i][b];
      float wr = swr[i][o], wi2 = swi[i][o];
      ar += xr * wr - xi * wi2;
      ai += xr * wi2 + xi * wr;
    }
    size_t dst = ((size_t)(b * 64 + o) * 128 + rio) * 64 + Y;
    yre[dst] = ar;
    yim[dst] = ai;
  }
}

// ---------------- mixing: LoRA experts (rank-4, gated) ----------------
__global__ __launch_bounds__(256) void k_mix_lora(const float* __restrict__ xre,
                                                  const float* __restrict__ xim,
                                                  const float* __restrict__ la1re, const float* __restrict__ la1im,
                                                  const float* __restrict__ lb1re, const float* __restrict__ lb1im,
                                                  const float* __restrict__ la2re, const float* __restrict__ la2im,
                                                  const float* __restrict__ lb2re, const float* __restrict__ lb2im,
                                                  const float* __restrict__ gate1, const float* __restrict__ gate2,
                                                  float* __restrict__ yre, float* __restrict__ yim) {
  int e = blockIdx.x;        // 0..14
  int corner = blockIdx.y;   // 0 top / 1 bottom
  int RR = (e + 1) >> 2, CC = (e + 1) & 3;
  int row0 = corner ? (64 + (3 - RR) * 16) : (RR * 16);
  int col0 = CC * 16;
  const float* lare = corner ? la2re : la1re;
  const float* laim = corner ? la2im : la1im;
  const float* lbre = corner ? lb2re : lb1re;
  const float* lbim = corner ? lb2im : lb1im;
  const float* gate = corner ? gate2 : gate1;
  float g = 0.1f / (1.0f + expf(-gate[e]));   // SCALING * sigmoid(gate)

  __shared__ float slar[4][64], slai[4][64];
  __shared__ float szr[4][4][256], szi[4][4][256];  // [b-half][rank][pos]
  int t = threadIdx.x;
  { int r = t >> 6, i = t & 63;
    slar[r][i] = lare[(size_t)e * 256 + t];
    slai[r][i] = laim[(size_t)e * 256 + t]; }
  __syncthreads();

  for (int half = 0; half < 2; ++half) {
    // z[b,r,pos] = sum_i la[r,i] * Xtile[b,i,pos]
    for (int q = t; q < 1024; q += 256) {
      int bl = q >> 8, pos = q & 255;
      int b = half * 4 + bl;
      int Xp = pos >> 4, Yp = pos & 15;
      size_t base = ((size_t)(b * 64) * 128 + (row0 + Xp)) * 64 + (col0 + Yp);
      float zr[4] = {0, 0, 0, 0}, zi[4] = {0, 0, 0, 0};
      for (int i = 0; i < 64; ++i) {
        size_t src = base + (size_t)i * 8192;
        float xr = xre[src], xi = xim[src];
        for (int r = 0; r < 4; ++r) {
          zr[r] += xr * slar[r][i] - xi * slai[r][i];
          zi[r] += xr * slai[r][i] + xi * slar[r][i];
        }
      }
      for (int r = 0; r < 4; ++r) { szr[bl][r][pos] = zr[r]; szi[bl][r][pos] = zi[r]; }
    }
    __syncthreads();
    // out[b,o,pos] = g * sum_r lb[o,r,pos] * z[b,r,pos]
    for (int q = t; q < 16384; q += 256) {
      int o = q >> 8, pos = q & 255;
      int Xp = pos >> 4, Yp = pos & 15;
      int ro = row0 + Xp, ky = col0 + Yp;
      float lbr[4], lbi[4];
      for (int r = 0; r < 4; ++r) {
        size_t src = (((size_t)(e * 64 + o) * 4 + r) * 16 + Xp) * 16 + Yp;
        lbr[r] = lbre[src];
        lbi[r] = lbim[src];
      }
      for (int bl = 0; bl < 4; ++bl) {
        int b = half * 4 + bl;
        float ar = 0.f, ai = 0.f;
        for (int r = 0; r < 4; ++r) {
          float zr = szr[bl][r][pos], zi = szi[bl][r][pos];
          ar += zr * lbr[r] - zi * lbi[r];
          ai += zr * lbi[r] + zi * lbr[r];
        }
        size_t dst = ((size_t)(b * 64 + o) * 128 + ro) * 64 + ky;
        yre[dst] = ar * g;
        yim[dst] = ai * g;
      }
    }
    __syncthreads();
  }
}

// ---------------- stage I1: inverse DFT along H ----------------
// U[h][ky] = sum_r e^{+2pi i kx h/256} Y[r][ky]; Ure = cos@Yre - sin@Yim ; Uim = sin@Yre + cos@Yim
__global__ __launch_bounds__(256) void k_idft_h(const float* __restrict__ yre,
                                                const float* __restrict__ yim,
                                                const float* __restrict__ ghc,
                                                const float* __restrict__ ghs,
                                                const float* __restrict__ ghms,
                                                float* __restrict__ ure,
                                                float* __restrict__ uim) {
  int img  = blockIdx.x;
  int tile = blockIdx.y * 8 + (threadIdx.x >> 5); // 0..63
  int mt = tile >> 2, nt = tile & 3;
  int lane = threadIdx.x & 31;
  int ln = lane & 15;
  int kh = (lane >> 4) << 1;
  int m = mt * 16 + ln;   // h
  int n = nt * 16 + ln;   // ky
  size_t yb = (size_t)img * 8192;
  v8f ar = {0,0,0,0,0,0,0,0}, ai = {0,0,0,0,0,0,0,0};
  for (int k0 = 0; k0 < 128; k0 += 4) {
    v2f ac;  ac.x  = ghc[m * 128 + k0 + kh];  ac.y  = ghc[m * 128 + k0 + kh + 1];
    v2f as;  as.x  = ghs[m * 128 + k0 + kh];  as.y  = ghs[m * 128 + k0 + kh + 1];
    v2f ams; ams.x = ghms[m * 128 + k0 + kh]; ams.y = ghms[m * 128 + k0 + kh + 1];
    v2f br;  br.x  = yre[yb + (size_t)(k0 + kh) * 64 + n]; br.y = yre[yb + (size_t)(k0 + kh + 1) * 64 + n];
    v2f bi;  bi.x  = yim[yb + (size_t)(k0 + kh) * 64 + n]; bi.y = yim[yb + (size_t)(k0 + kh + 1) * 64 + n];
    ar = wmma4(ac, br, ar);
    ar = wmma4(ams, bi, ar);
    ai = wmma4(as, br, ai);
    ai = wmma4(ac, bi, ai);
  }
  size_t ub = (size_t)img * 16384;
  int rowoff = mt * 16 + ((lane >> 4) << 3);
  for (int j = 0; j < 8; ++j) {
    ure[ub + (size_t)(rowoff + j) * 64 + n] = ar[j];
    uim[ub + (size_t)(rowoff + j) * 64 + n] = ai[j];
  }
}

// ---------------- stage I2: inverse DFT along W (Hermitian fold, real out) ----------------
// y[h][w] = Ure[h][:] @ Gwr[:, w] + Uim[h][:] @ Gwi[:, w]
__global__ __launch_bounds__(256) void k_idft_w(const float* __restrict__ ure,
                                                const float* __restrict__ uim,
                                                const float* __restrict__ gwr,
                                                const float* __restrict__ gwi,
                                                float* __restrict__ y) {
  int img  = blockIdx.x;
  int tile = blockIdx.y * 8 + (threadIdx.x >> 5); // 0..255
  int mt = tile >> 4, nt = tile & 15;
  int lane = threadIdx.x & 31;
  int ln = lane & 15;
  int kh = (lane >> 4) << 1;
  int m = mt * 16 + ln;   // h
  int n = nt * 16 + ln;   // w
  size_t ub = (size_t)img * 16384;
  v8f acc = {0,0,0,0,0,0,0,0};
  for (int k0 = 0; k0 < 64; k0 += 4) {
    v2f arr; arr.x = ure[ub + (size_t)m * 64 + k0 + kh]; arr.y = ure[ub + (size_t)m * 64 + k0 + kh + 1];
    v2f aii; aii.x = uim[ub + (size_t)m * 64 + k0 + kh]; aii.y = uim[ub + (size_t)m * 64 + k0 + kh + 1];
    v2f brr; brr.x = gwr[(k0 + kh) * 256 + n];           brr.y = gwr[(k0 + kh + 1) * 256 + n];
    v2f bii; bii.x = gwi[(k0 + kh) * 256 + n];           bii.y = gwi[(k0 + kh + 1) * 256 + n];
    acc = wmma4(arr, brr, acc);
    acc = wmma4(aii, bii, acc);
  }
  size_t yb = (size_t)img * 65536;
  int rowoff = mt * 16 + ((lane >> 4) << 3);
  for (int j = 0; j < 8; ++j) {
    y[yb + (size_t)(rowoff + j) * 256 + n] = acc[j];
  }
}

// ---------------- sploss scalar ----------------
__global__ void k_sploss(const float* __restrict__ g1, const float* __restrict__ g2,
                         float* __restrict__ out, int idx) {
  if (threadIdx.x == 0 && blockIdx.x == 0) {
    float s1 = 0.f, s2 = 0.f;
    for (int e = 0; e < 15; ++e) {
      s1 += 1.0f / (1.0f + expf(-g1[e]));
      s2 += 1.0f / (1.0f + expf(-g2[e]));
    }
    out[idx] = s1 / 15.0f + s2 / 15.0f;
  }
}

extern "C" void kernel_launch(void* const* d_in, const int* in_sizes, int n_in,
                              void* d_out, int out_size, void* d_ws, size_t ws_size,
                              hipStream_t stream) {
  (void)in_sizes; (void)n_in; (void)ws_size;
  const float* x     = (const float*)d_in[0];
  const float* w1re  = (const float*)d_in[1];
  const float* w1im  = (const float*)d_in[2];
  const float* w2re  = (const float*)d_in[3];
  const float* w2im  = (const float*)d_in[4];
  const float* la1re = (const float*)d_in[5];
  const float* la1im = (const float*)d_in[6];
  const float* lb1re = (const float*)d_in[7];
  const float* lb1im = (const float*)d_in[8];
  const float* la2re = (const float*)d_in[9];
  const float* la2im = (const float*)d_in[10];
  const float* lb2re = (const float*)d_in[11];
  const float* lb2im = (const float*)d_in[12];
  const float* gate1 = (const float*)d_in[13];
  const float* gate2 = (const float*)d_in[14];
  float* out = (float*)d_out;
  float* ws  = (float*)d_ws;

  k_twiddle<<<128, 256, 0, stream>>>(ws);
  k_dft_w<<<dim3(512, 8), 256, 0, stream>>>(x, ws + OFF_FWC, ws + OFF_FWS,
                                            ws + OFF_TRE, ws + OFF_TIM);
  k_dft_h<<<dim3(512, 4), 256, 0, stream>>>(ws + OFF_TRE, ws + OFF_TIM,
                                            ws + OFF_FHC, ws + OFF_FHS, ws + OFF_FHMS,
                                            ws + OFF_XRE, ws + OFF_XIM);
  k_mix_base<<<dim3(256, 2), 256, 0, stream>>>(ws + OFF_XRE, ws + OFF_XIM,
                                               w1re, w1im, w2re, w2im,
                                               ws + OFF_YRE, ws + OFF_YIM);
  k_mix_lora<<<dim3(15, 2), 256, 0, stream>>>(ws + OFF_XRE, ws + OFF_XIM,
                                              la1re, la1im, lb1re, lb1im,
                                              la2re, la2im, lb2re, lb2im,
                                              gate1, gate2,
                                              ws + OFF_YRE, ws + OFF_YIM);
  // U aliases T (T is dead after k_dft_h)
  k_idft_h<<<dim3(512, 8), 256, 0, stream>>>(ws + OFF_YRE, ws + OFF_YIM,
                                             ws + OFF_GHC, ws + OFF_GHS, ws + OFF_GHMS,
                                             ws + OFF_TRE, ws + OFF_TIM);
  k_idft_w<<<dim3(512, 32), 256, 0, stream>>>(ws + OFF_TRE, ws + OFF_TIM,
                                              ws + OFF_GWR, ws + OFF_GWI, out);
  k_sploss<<<1, 32, 0, stream>>>(gate1, gate2, out, out_size - 1);
}